// VectorQuantizer_36301063586577
// MI455X (gfx1250) — compile-verified
//
#include <hip/hip_runtime.h>
#include <hip/hip_bf16.h>

typedef __attribute__((ext_vector_type(2))) float v2f;
typedef __attribute__((ext_vector_type(4))) float v4f;
typedef __attribute__((ext_vector_type(8))) float v8f;

#define KCODES   1024
#define DDIM     64
#define NROWS    32768          // B*T = 8*4096
#define ROWS_PB  16
#define NBLOCKS  (NROWS / ROWS_PB)   // 2048
#define WAVES    8
#define COLS_PW  (KCODES / WAVES)    // 128
#define NT_PW    (COLS_PW / 16)      // 8

// ---------------------------------------------------------------------------
// Kernel 0: codebook column squared norms  enorm[k] = sum_d E[d,k]^2
// ---------------------------------------------------------------------------
__global__ __launch_bounds__(256) void vq_enorm_kernel(const float* __restrict__ E,
                                                       float* __restrict__ enorm) {
    int k = blockIdx.x * blockDim.x + threadIdx.x;
    if (k < KCODES) {
        float s = 0.f;
#pragma unroll
        for (int d = 0; d < DDIM; ++d) {
            float e = E[(size_t)d * KCODES + k];
            s += e * e;
        }
        enorm[k] = s;
    }
}

// ---------------------------------------------------------------------------
// Main kernel: 1 block = 16 rows of x.
//  - sim tile via v_wmma_f32_16x16x4_f32 (fp32, exact-ish vs reference)
//  - argmin over 1024 codes with reference tie-breaking (lowest index)
//  - gather quantized, loss partial, stream one-hot encodings (NT stores)
// ---------------------------------------------------------------------------
__global__ __launch_bounds__(256) void vq_main_kernel(const float* __restrict__ x,
                                                      const float* __restrict__ E,
                                                      const float* __restrict__ enorm,
                                                      float* __restrict__ out_q,
                                                      float* __restrict__ out_enc,
                                                      float* __restrict__ out_idx,
                                                      float* __restrict__ loss_partial) {
    __shared__ float s_minv[WAVES][ROWS_PB];
    __shared__ int   s_mini[WAVES][ROWS_PB];
    __shared__ int   s_idx[ROWS_PB];
    __shared__ float s_red[256];

    const int tid  = threadIdx.x;
    const int wave = tid >> 5;
    const int lane = tid & 31;
    const int l15  = lane & 15;
    const int half = lane >> 4;              // 0: K={0,1}, 1: K={2,3} within each K4 chunk
    const int row0 = blockIdx.x * ROWS_PB;

    // ---- A fragments: A[m=l15][k = 4*kc + 2*half + {0,1}]  (ISA 32-bit A 16x4 layout)
    v2f afrag[16];
    {
        const float* xrow = x + (size_t)(row0 + l15) * DDIM + 2 * half;
#pragma unroll
        for (int kc = 0; kc < 16; ++kc)
            afrag[kc] = *(const v2f*)(xrow + 4 * kc);
    }

    // ---- sweep this wave's 128 columns in 16-wide tiles, track per-lane argmin
    float minv[8];
    int   mini[8];
#pragma unroll
    for (int i = 0; i < 8; ++i) { minv[i] = 3.402823e38f; mini[i] = 0x7fffffff; }

    const int colbase = wave * COLS_PW;
    for (int nt = 0; nt < NT_PW; ++nt) {
        const int col = colbase + nt * 16 + l15;
        v8f acc = {0.f, 0.f, 0.f, 0.f, 0.f, 0.f, 0.f, 0.f};
        const float* ecol = E + col;
#pragma unroll
        for (int kc = 0; kc < 16; ++kc) {
            // B fragment: B[k = 4*kc + 2*half + {0,1}][n = col]
            const int kb = 4 * kc + 2 * half;
            v2f bfrag;
            bfrag.x = ecol[(size_t)kb * KCODES];
            bfrag.y = ecol[(size_t)(kb + 1) * KCODES];
            acc = __builtin_amdgcn_wmma_f32_16x16x4_f32(
                false, afrag[kc], false, bfrag, (short)0, acc, false, false);
        }
        // dist (up to the row-constant ||x||^2) = ||e||^2 - 2*sim
        const float en = enorm[col];
#pragma unroll
        for (int i = 0; i < 8; ++i) {
            float d = en - 2.0f * acc[i];          // row m = i + 8*half, col = col
            if (d < minv[i]) { minv[i] = d; mini[i] = col; }
        }
    }

    // ---- reduce across the 16 lanes sharing each row (xor stays within halves)
#pragma unroll
    for (int m = 8; m >= 1; m >>= 1) {
#pragma unroll
        for (int i = 0; i < 8; ++i) {
            float ov = __shfl_xor(minv[i], m, 32);
            int   oi = __shfl_xor(mini[i], m, 32);
            if (ov < minv[i] || (ov == minv[i] && oi < mini[i])) {
                minv[i] = ov; mini[i] = oi;
            }
        }
    }
    if (l15 == 0) {
#pragma unroll
        for (int i = 0; i < 8; ++i) {
            s_minv[wave][half * 8 + i] = minv[i];
            s_mini[wave][half * 8 + i] = mini[i];
        }
    }
    __syncthreads();

    // ---- final per-row argmin across waves (fixed order -> deterministic)
    if (tid < ROWS_PB) {
        float bv = s_minv[0][tid];
        int   bi = s_mini[0][tid];
#pragma unroll
        for (int w = 1; w < WAVES; ++w) {
            float v = s_minv[w][tid];
            int   i = s_mini[w][tid];
            if (v < bv || (v == bv && i < bi)) { bv = v; bi = i; }
        }
        s_idx[tid] = bi;
        out_idx[row0 + tid] = (float)bi;
    }
    __syncthreads();

    // ---- quantized gather + commitment-loss partial (16*64 = 1024 elems, 4/thread)
    float lsum = 0.f;
#pragma unroll
    for (int j = 0; j < 4; ++j) {
        const int e = tid + j * 256;
        const int r = e >> 6;
        const int d = e & 63;
        const int idx = s_idx[r];
        const float q  = E[(size_t)d * KCODES + idx];
        const float xv = x[(size_t)(row0 + r) * DDIM + d];
        const float df = q - xv;
        lsum += df * df;
        out_q[(size_t)(row0 + r) * DDIM + d] = q;
    }
    s_red[tid] = lsum;
    __syncthreads();
#pragma unroll
    for (int s = 128; s > 0; s >>= 1) {
        if (tid < s) s_red[tid] += s_red[tid + s];
        __syncthreads();
    }
    if (tid == 0) loss_partial[blockIdx.x] = s_red[0];

    // ---- stream one-hot encodings: 16 rows x 1024 = 4096 float4 NT stores
    float* enc_blk = out_enc + (size_t)row0 * KCODES;
#pragma unroll
    for (int j = 0; j < 16; ++j) {
        const int c   = tid + j * 256;   // float4 chunk id within the 16x1024 slab
        const int p   = c << 2;          // float offset
        const int r   = p >> 10;         // local row
        const int col = p & 1023;
        const int idx = s_idx[r];
        v4f v;
        v.x = (idx == col)     ? 1.f : 0.f;
        v.y = (idx == col + 1) ? 1.f : 0.f;
        v.z = (idx == col + 2) ? 1.f : 0.f;
        v.w = (idx == col + 3) ? 1.f : 0.f;
        __builtin_nontemporal_store(v, (v4f*)(enc_blk + p));
    }
}

// ---------------------------------------------------------------------------
// Final loss reduction (single block, fixed order -> bitwise deterministic)
// ---------------------------------------------------------------------------
__global__ __launch_bounds__(256) void vq_loss_kernel(const float* __restrict__ part,
                                                      float* __restrict__ out_loss) {
    __shared__ float s[256];
    float t = 0.f;
    for (int i = threadIdx.x; i < NBLOCKS; i += 256) t += part[i];
    s[threadIdx.x] = t;
    __syncthreads();
#pragma unroll
    for (int k = 128; k > 0; k >>= 1) {
        if (threadIdx.x < k) s[threadIdx.x] += s[threadIdx.x + k];
        __syncthreads();
    }
    if (threadIdx.x == 0)
        *out_loss = s[0] * (1.0f / (float)(NROWS * DDIM));
}

// ---------------------------------------------------------------------------
extern "C" void kernel_launch(void* const* d_in, const int* in_sizes, int n_in,
                              void* d_out, int out_size, void* d_ws, size_t ws_size,
                              hipStream_t stream) {
    (void)in_sizes; (void)n_in; (void)out_size; (void)ws_size;
    const float* x = (const float*)d_in[0];       // [8,4096,64]
    const float* E = (const float*)d_in[1];       // [64,1024]

    float* out      = (float*)d_out;
    float* out_q    = out;                                    // 2,097,152
    float* out_enc  = out_q + (size_t)NROWS * DDIM;           // 33,554,432
    float* out_idx  = out_enc + (size_t)NROWS * KCODES;       // 32,768
    float* out_loss = out_idx + NROWS;                        // 1

    float* enorm    = (float*)d_ws;                           // 1024 floats
    float* partials = enorm + KCODES;                         // 2048 floats

    vq_enorm_kernel<<<(KCODES + 255) / 256, 256, 0, stream>>>(E, enorm);
    vq_main_kernel<<<NBLOCKS, 256, 0, stream>>>(x, E, enorm,
                                                out_q, out_enc, out_idx, partials);
    vq_loss_kernel<<<1, 256, 0, stream>>>(partials, out_loss);
}